// CrossModalAttention_50946902065329
// MI455X (gfx1250) — compile-verified
//
#include <hip/hip_runtime.h>
#include <hip/hip_bf16.h>

typedef float v2f __attribute__((ext_vector_type(2)));
typedef float v8f __attribute__((ext_vector_type(8)));

#define S_ 64
#define P_ 2048
#define C_ 32
#define A_ 2048
#define D_ 128
#define E_ 64

// One wave computes a 16x16 D tile of D = A[K] @ B[K x LDB] + bias, f32 WMMA K=4 steps.
// A: row-major [*, K] starting at tile row. B: row-major [K, LDB]. D: row-major, stride ldd.
template<int K, int LDB>
__device__ __forceinline__ void wmma_tile_f32(const float* __restrict__ Arow,
                                              const float* __restrict__ B,
                                              const float* __restrict__ bias,
                                              float* __restrict__ Drow,
                                              int n0, int ldd)
{
    const int lane = threadIdx.x & 31;
    const int mrow = lane & 15;           // A row within tile
    const int koff = (lane >> 4) * 2;     // K sub-offset for upper half-wave
    const int ncol = n0 + (lane & 15);    // B/D column

    v8f c = {};
#pragma unroll
    for (int k = 0; k < K; k += 4) {
        v2f a, b;
        a.x = Arow[(size_t)mrow * K + k + koff];
        a.y = Arow[(size_t)mrow * K + k + koff + 1];
        b.x = B[(size_t)(k + koff) * LDB + ncol];
        b.y = B[(size_t)(k + koff + 1) * LDB + ncol];
        // 8 args: (neg_a, A, neg_b, B, c_mod, C, reuse_a, reuse_b)
        c = __builtin_amdgcn_wmma_f32_16x16x4_f32(false, a, false, b, (short)0, c, false, false);
    }

    const float bv = bias[ncol];
    const int mbase = (lane >> 4) * 8;
#pragma unroll
    for (int r = 0; r < 8; ++r)
        Drow[(size_t)(mbase + r) * ldd + ncol] = c[r] + bv;
}

// att1[s,p,e] = gs[s,p,:] @ W_sn + b_sn  ->  GEMM M=S*P, K=32, N=64
__launch_bounds__(256)
__global__ void att1_kernel(const float* __restrict__ gs, const float* __restrict__ W,
                            const float* __restrict__ b, float* __restrict__ att1)
{
    const int wave  = threadIdx.x >> 5;
    const int mtile = blockIdx.x * 8 + wave;     // 8192 tiles of 16 rows
    const int n0    = blockIdx.y * 16;
    wmma_tile_f32<C_, E_>(gs + (size_t)mtile * 16 * C_, W, b,
                          att1 + (size_t)mtile * 16 * E_, n0, E_);
}

// att2[a,e] = dyn[a,:] @ W_df + b_df  ->  GEMM M=2048, K=128, N=64
__launch_bounds__(256)
__global__ void att2_kernel(const float* __restrict__ dyn, const float* __restrict__ W,
                            const float* __restrict__ b, float* __restrict__ att2)
{
    const int wave  = threadIdx.x >> 5;
    const int mtile = blockIdx.x * 8 + wave;     // 128 tiles of 16 rows
    const int n0    = blockIdx.y * 16;
    wmma_tile_f32<D_, E_>(dyn + (size_t)mtile * 16 * D_, W, b,
                          att2 + (size_t)mtile * 16 * E_, n0, E_);
}

// Per-agent fused: logits -> softmax over P -> alpha-weighted pooling of gs[s].
__launch_bounds__(256)
__global__ void fuse_kernel(const float* __restrict__ att1, const float* __restrict__ att2,
                            const int* __restrict__ scene_idx, const float* __restrict__ gs,
                            const float* __restrict__ w_fc, const float* __restrict__ b_fc,
                            float* __restrict__ out)
{
    __shared__ float att2_s[E_];
    __shared__ float wfc_s[E_];
    __shared__ float lse[P_];        // exp(logit - max) per pixel
    __shared__ float red[8];         // per-wave reductions
    __shared__ float pool[8][C_];

    const int a   = blockIdx.x;
    const int tid = threadIdx.x;
    const int s   = scene_idx[a];

    if (tid < E_) {
        att2_s[tid] = att2[(size_t)a * E_ + tid];
        wfc_s[tid]  = w_fc[tid];
    }
    __syncthreads();

    const float bfc = b_fc[0];
    const float* base = att1 + (size_t)s * P_ * E_;

    // ---- logits for 8 pixels per thread ----
    float lg[8];
    float lmax = -INFINITY;
#pragma unroll
    for (int i = 0; i < 8; ++i) {
        const int p = tid + i * 256;
        const float4* ap = (const float4*)(base + (size_t)p * E_);
        float acc = bfc;
#pragma unroll
        for (int e4 = 0; e4 < E_ / 4; ++e4) {
            const float4 v = ap[e4];
            const int e = e4 * 4;
            acc += wfc_s[e + 0] * fmaxf(v.x + att2_s[e + 0], 0.0f);
            acc += wfc_s[e + 1] * fmaxf(v.y + att2_s[e + 1], 0.0f);
            acc += wfc_s[e + 2] * fmaxf(v.z + att2_s[e + 2], 0.0f);
            acc += wfc_s[e + 3] * fmaxf(v.w + att2_s[e + 3], 0.0f);
        }
        lg[i] = acc;
        lmax = fmaxf(lmax, acc);
    }

    // ---- block max ----
#pragma unroll
    for (int off = 16; off; off >>= 1)
        lmax = fmaxf(lmax, __shfl_xor(lmax, off, 32));
    if ((tid & 31) == 0) red[tid >> 5] = lmax;
    __syncthreads();
    float bmax = red[0];
#pragma unroll
    for (int w = 1; w < 8; ++w) bmax = fmaxf(bmax, red[w]);

    // ---- exp + block sum ----
    float lsum = 0.0f;
#pragma unroll
    for (int i = 0; i < 8; ++i) {
        const float e = __expf(lg[i] - bmax);
        lse[tid + i * 256] = e;
        lsum += e;
    }
#pragma unroll
    for (int off = 16; off; off >>= 1)
        lsum += __shfl_xor(lsum, off, 32);
    __syncthreads();                  // red reads done; lse writes visible after next sync
    if ((tid & 31) == 0) red[tid >> 5] = lsum;
    __syncthreads();
    float bsum = red[0];
#pragma unroll
    for (int w = 1; w < 8; ++w) bsum += red[w];
    const float inv = 1.0f / bsum;

    // ---- alpha-weighted pooling: lanes 0..31 <-> channels (coalesced 128B rows) ----
    const int c = tid & 31;
    const int g = tid >> 5;           // 8 pixel groups of 256
    const float* gsrow = gs + ((size_t)s * P_ + (size_t)g * 256) * C_ + c;
    float partial = 0.0f;
    for (int j = 0; j < 256; ++j)
        partial += lse[g * 256 + j] * gsrow[(size_t)j * C_];
    pool[g][c] = partial;
    __syncthreads();

    if (tid < C_) {
        float acc = 0.0f;
#pragma unroll
        for (int w = 0; w < 8; ++w) acc += pool[w][tid];
        out[(size_t)a * C_ + tid] = inv * acc;
    }
}

extern "C" void kernel_launch(void* const* d_in, const int* in_sizes, int n_in,
                              void* d_out, int out_size, void* d_ws, size_t ws_size,
                              hipStream_t stream)
{
    const float* gs   = (const float*)d_in[0];   // [S,P,C]
    const int*   sidx = (const int*)  d_in[1];   // [A]
    const float* dyn  = (const float*)d_in[2];   // [A,D]
    const float* W_sn = (const float*)d_in[3];   // [C,E]
    const float* b_sn = (const float*)d_in[4];   // [E]
    const float* W_df = (const float*)d_in[5];   // [D,E]
    const float* b_df = (const float*)d_in[6];   // [E]
    const float* w_fc = (const float*)d_in[7];   // [E]
    const float* b_fc = (const float*)d_in[8];   // [1]
    float*       out  = (float*)d_out;           // [A,C]

    float* att1 = (float*)d_ws;                              // S*P*E = 8.4M floats
    float* att2 = att1 + (size_t)S_ * P_ * E_;               // A*E   = 131K floats

    // Stage 1: att1 = gs @ W_sn + b_sn   (M=131072, K=32, N=64)
    {
        dim3 grid((S_ * P_ / 16) / 8, E_ / 16);   // 1024 x 4
        att1_kernel<<<grid, 256, 0, stream>>>(gs, W_sn, b_sn, att1);
    }
    // Stage 2: att2 = dyn @ W_df + b_df  (M=2048, K=128, N=64)
    {
        dim3 grid((A_ / 16) / 8, E_ / 16);        // 16 x 4
        att2_kernel<<<grid, 256, 0, stream>>>(dyn, W_df, b_df, att2);
    }
    // Stage 3: fused gather/relu/logit/softmax/pool, one block per agent
    fuse_kernel<<<A_, 256, 0, stream>>>(att1, att2, sidx, gs, w_fc, b_fc, out);
}